// loss_1108101563123
// MI455X (gfx1250) — compile-verified
//
#include <hip/hip_runtime.h>
#include <hip/hip_bf16.h>
#include <math.h>

typedef __attribute__((ext_vector_type(2))) float v2f;
typedef __attribute__((ext_vector_type(8))) float v8f;

#define NUM_CITIES 88
#define NUM_YEARS  6
#define N_ROWS     200000
#define G_ELEMS    (N_ROWS * NUM_CITIES)   // 17,600,000
#define G_VEC4     (G_ELEMS / 4)           // 4,400,000
#define UO_ELEMS   (N_ROWS * NUM_YEARS)    // 1,200,000

#define MAIN_BLOCK 352                     // 11 waves; 352*4 = 1408 = 16*88 elems/block-iter
#define MAIN_GRID  880                     // stride 880*1408 elems, multiple of 88

// ws layout (floats): ws[0] = scalar accumulator, ws[1..88] = column L1 sums of |G|
__global__ void zero_ws_kernel(float* ws) {
    int i = threadIdx.x;
    if (i < 1 + NUM_CITIES) ws[i] = 0.0f;
}

__global__ __launch_bounds__(MAIN_BLOCK)
void gdw_kernel(const float* __restrict__ G, const float* __restrict__ D,
                const float* __restrict__ W, float* __restrict__ ws) {
    __shared__ float s_col[NUM_CITIES];
    __shared__ float s_wave[MAIN_BLOCK / 32];

    const int tid = threadIdx.x;
    if (tid < NUM_CITIES) s_col[tid] = 0.0f;
    __syncthreads();

    const int tid0    = blockIdx.x * MAIN_BLOCK + tid;   // float4 index of first iter
    const int stride4 = MAIN_GRID * MAIN_BLOCK;          // float4 stride (mult. of 22 -> *4 mult of 88)

    // fixed columns for this thread's 4 vector lanes (stride*4 % 88 == 0)
    int c0 = (tid0 * 4) % NUM_CITIES;
    int c1 = c0 + 1; if (c1 >= NUM_CITIES) c1 -= NUM_CITIES;
    int c2 = c0 + 2; if (c2 >= NUM_CITIES) c2 -= NUM_CITIES;
    int c3 = c0 + 3; if (c3 >= NUM_CITIES) c3 -= NUM_CITIES;

    const float4* __restrict__ G4 = (const float4*)G;
    const float4* __restrict__ D4 = (const float4*)D;
    const float4* __restrict__ W4 = (const float4*)W;

    float a0 = 0.f, a1 = 0.f, a2 = 0.f, a3 = 0.f;   // per-column |G| partials
    float sacc = 0.f;                               // combined scalar partial

    for (int i = tid0; i < G_VEC4; i += stride4) {
        float4 g = G4[i];
        float4 d = D4[i];
        float4 v = W4[i];

        float r;
        r = fmaf(-v.x, d.x, g.x);
        sacc += 100.0f * r * r + 0.01f * v.x * v.x + 100.0f * fminf(g.x, 0.0f);
        a0 += fabsf(g.x);
        r = fmaf(-v.y, d.y, g.y);
        sacc += 100.0f * r * r + 0.01f * v.y * v.y + 100.0f * fminf(g.y, 0.0f);
        a1 += fabsf(g.y);
        r = fmaf(-v.z, d.z, g.z);
        sacc += 100.0f * r * r + 0.01f * v.z * v.z + 100.0f * fminf(g.z, 0.0f);
        a2 += fabsf(g.z);
        r = fmaf(-v.w, d.w, g.w);
        sacc += 100.0f * r * r + 0.01f * v.w * v.w + 100.0f * fminf(g.w, 0.0f);
        a3 += fabsf(g.w);
    }

    // block-level column combine via LDS float atomics (ds_add_f32)
    atomicAdd(&s_col[c0], a0);
    atomicAdd(&s_col[c1], a1);
    atomicAdd(&s_col[c2], a2);
    atomicAdd(&s_col[c3], a3);

    // wave32 tree-reduce of scalar partial
    for (int m = 16; m > 0; m >>= 1) sacc += __shfl_xor(sacc, m);
    if ((tid & 31) == 0) s_wave[tid >> 5] = sacc;
    __syncthreads();

    if (tid == 0) {
        float t = 0.f;
        #pragma unroll
        for (int k = 0; k < MAIN_BLOCK / 32; ++k) t += s_wave[k];
        atomicAdd(&ws[0], t);
    }
    if (tid < NUM_CITIES) atomicAdd(&ws[1 + tid], s_col[tid]);
}

__global__ void uo_kernel(const float* __restrict__ U, const float* __restrict__ O,
                          float* __restrict__ ws) {
    int i = blockIdx.x * blockDim.x + threadIdx.x;
    const int stride = gridDim.x * blockDim.x;
    float acc = 0.f;
    for (; i < UO_ELEMS; i += stride) {
        int col = i % NUM_YEARS;
        if (col != NUM_YEARS - 1) {
            float df = U[i] - O[i];
            acc += df * df;
        }
    }
    for (int m = 16; m > 0; m >>= 1) acc += __shfl_xor(acc, m);
    if ((threadIdx.x & 31) == 0) atomicAdd(&ws[0], acc);
}

// One wave: 88 logs, then WMMA ones-trick reduction (C[m,n] = sum_k B[k,n];
// B.v0 = partial, B.v1 = 0 => every lane's C[0] = p[n]+p[n+16]), finish with
// 4 xor shuffles.
__global__ void finalize_kernel(const float* __restrict__ ws, float* __restrict__ out) {
    const int lane = threadIdx.x;   // blockDim = 32, one full wave, EXEC all 1s at the WMMA
    float p = 0.f;
    for (int c = lane; c < NUM_CITIES; c += 32) p += logf(ws[1 + c]);

    v2f a; a[0] = 1.0f; a[1] = 1.0f;   // A = 16x4 all-ones
    v2f b; b[0] = p;    b[1] = 0.0f;   // B rows K0/K2 = partials, K1/K3 = 0
    v8f c = {};
    c = __builtin_amdgcn_wmma_f32_16x16x4_f32(
        /*neg_a=*/false, a, /*neg_b=*/false, b,
        /*c_mod=*/(short)0, c, /*reuse_a=*/false, /*reuse_b=*/false);

    float t = c[0];                    // p[lane&15] + p[(lane&15)+16]
    t += __shfl_xor(t, 1);
    t += __shfl_xor(t, 2);
    t += __shfl_xor(t, 4);
    t += __shfl_xor(t, 8);

    if (lane == 0) out[0] = ws[0] + 1e-4f * t;   // stddeve^2 * sum(log colL1)
}

extern "C" void kernel_launch(void* const* d_in, const int* in_sizes, int n_in,
                              void* d_out, int out_size, void* d_ws, size_t ws_size,
                              hipStream_t stream) {
    (void)in_sizes; (void)n_in; (void)out_size; (void)ws_size;
    const float* G = (const float*)d_in[0];
    const float* O = (const float*)d_in[1];   // "out" observation matrix
    const float* U = (const float*)d_in[2];
    // d_in[3] = V, unused by the reference
    const float* D = (const float*)d_in[4];
    const float* W = (const float*)d_in[5];
    float* ws   = (float*)d_ws;
    float* loss = (float*)d_out;

    zero_ws_kernel<<<1, 128, 0, stream>>>(ws);
    gdw_kernel<<<MAIN_GRID, MAIN_BLOCK, 0, stream>>>(G, D, W, ws);
    uo_kernel<<<512, 256, 0, stream>>>(U, O, ws);
    finalize_kernel<<<1, 32, 0, stream>>>(ws, loss);
}